// RecurrentKernel_80135499808800
// MI455X (gfx1250) — compile-verified
//
#include <hip/hip_runtime.h>
#include <math.h>

typedef float v2f __attribute__((ext_vector_type(2)));
typedef float v8f __attribute__((ext_vector_type(8)));

#define N_ROWS   8191        // T-1
#define DDIM     128
#define LDG      8192        // gram leading dim in workspace
#define N_INIT   100
#define MOUT     8092        // N_ROWS - (N_INIT-1)
#define A_C      0.5f
#define B_C      0.9f
#define C_C      0.1f
#define EPS_C    1e-6f
#define RENORM_C 0.1f
#define TWO_OVER_PI 0.63661977236758134f

// ---------------------------------------------------------------------------
// Kernel 1: gram = X X^T, upper-triangle 16x16 tiles via V_WMMA_F32_16X16X4_F32
// One wave per tile. A and B tiles have identical lane layouts:
//   lane L<16 : row = base + L,     holds K = k0,   k0+1
//   lane L>=16: row = base + L-16,  holds K = k0+2, k0+3
// ---------------------------------------------------------------------------
__global__ void gram_wmma_kernel(const float* __restrict__ X,
                                 float* __restrict__ gram) {
    const int lane = threadIdx.x & 31;
    const int wave = threadIdx.x >> 5;
    const int tj   = blockIdx.x * 8 + wave;   // column tile
    const int ti   = blockIdx.y;              // row tile
    if (tj < ti) return;                      // upper triangle only (wave-uniform)

    const int r  = lane & 15;
    const int kh = (lane >> 4) * 2;           // 0 or 2

    const float* __restrict__ arow = X + (size_t)(ti * 16 + r) * DDIM + kh;
    const float* __restrict__ brow = X + (size_t)(tj * 16 + r) * DDIM + kh;

    v8f acc = {};
#pragma unroll 8
    for (int k0 = 0; k0 < DDIM; k0 += 4) {
        v2f a = *(const v2f*)(arow + k0);
        v2f b = *(const v2f*)(brow + k0);
        acc = __builtin_amdgcn_wmma_f32_16x16x4_f32(
            false, a, false, b, (short)0, acc, false, false);
    }

    // C/D layout: VGPR v, lanes 0-15 -> row v, lanes 16-31 -> row v+8; col = lane&15
    const int row0 = ti * 16 + (lane >> 4) * 8;
    const int col  = tj * 16 + (lane & 15);
#pragma unroll
    for (int v = 0; v < 8; ++v) {
        gram[(size_t)(row0 + v) * LDG + col] = acc[v];
    }
}

// ---------------------------------------------------------------------------
// Kernel 2: sequential diag scan -> uu, k_diag_final.  Single wave; lanes
// prefetch 32 diag values, chain runs uniformly on all lanes via __shfl.
// ---------------------------------------------------------------------------
__global__ void diag_scan_kernel(const float* __restrict__ gram,
                                 float* __restrict__ uu_ws,
                                 float* __restrict__ out_uu,
                                 float* __restrict__ out_kdiag_final) {
    const int lane = threadIdx.x;
    float kprev = 0.0f;
    for (int base = 0; base < N_ROWS; base += 32) {
        const int idx = base + lane;
        const float dval = (idx < N_ROWS) ? gram[(size_t)idx * (LDG + 1)] : 0.0f;
        float myuu = 0.0f;
        const int cnt = (N_ROWS - base < 32) ? (N_ROWS - base) : 32;
        for (int t = 0; t < cnt; ++t) {
            const float g  = __shfl(dval, t, 32);
            const float uv = A_C * g + B_C * kprev + C_C;
            // kernel_fn(uv, uv, uv)
            const float w    = 1.0f + 2.0f * uv;
            const float den  = sqrtf(w * w);
            float arg = 2.0f * uv / den;
            arg = fminf(fmaxf(arg, -1.0f + EPS_C), 1.0f - EPS_C);
            kprev = TWO_OVER_PI * asinf(arg);
            if (lane == t) myuu = uv;
        }
        if (idx < N_ROWS) {
            uu_ws[idx]  = myuu;
            out_uu[idx] = myuu;
        }
    }
    if (lane == 0) out_kdiag_final[0] = kprev;
}

// ---------------------------------------------------------------------------
// Kernel 3: one thread per upper diagonal d = j - i (0 <= d < N).
// k[s][s+d] = kernel_fn(A*g + B*k[s-1][s-1+d] + C, uu[s+d], uu[s]).
// Coalesced: lanes have consecutive d; at step s they touch gram[s*8193+d..],
// uu[s+d..] (contiguous) and uu[s] (scalar broadcast).
// Writes k_train upper triangle (+renorm) and k_final[N-1-d] at chain end.
// ---------------------------------------------------------------------------
__global__ void diag_chain_kernel(const float* __restrict__ gram,
                                  const float* __restrict__ uu,
                                  float* __restrict__ out_ktrain,
                                  float* __restrict__ out_kfinal) {
    const int d = blockIdx.x * blockDim.x + threadIdx.x;
    if (d >= N_ROWS) return;

    const int len = N_ROWS - d;
    float kprev = 0.0f;
    for (int s = 0; s < len; ++s) {
        const float g   = gram[(size_t)s * (LDG + 1) + d]; // gram[s][s+d]
        const float uuj = uu[s + d];
        const float uui = uu[s];
        const float uv  = A_C * g + B_C * kprev + C_C;
        const float prod = (1.0f + 2.0f * uuj) * (1.0f + 2.0f * uui);
        float arg = 2.0f * uv * rsqrtf(prod);
        arg = fminf(fmaxf(arg, -1.0f + EPS_C), 1.0f - EPS_C);
        kprev = TWO_OVER_PI * asinf(arg);
        if (s >= N_INIT - 1) { // i = s >= 99 implies j = s + d >= 99
            out_ktrain[(size_t)(s - (N_INIT - 1)) * MOUT + (s + d - (N_INIT - 1))] =
                kprev + RENORM_C * g;
        }
    }
    // chain end is (i = N-1-d, j = N-1); by symmetry k_final[N-1-d] = kprev
    out_kfinal[N_ROWS - 1 - d] = kprev;
}

// ---------------------------------------------------------------------------
// Kernel 4: mirror lower triangle of k_train from upper via LDS tile transpose.
// out[r][c] = out[c][r] for r > c.
// ---------------------------------------------------------------------------
__global__ void mirror_kernel(float* __restrict__ out) {
    const int bc = blockIdx.x;   // column tile of destination (lower)
    const int br = blockIdx.y;   // row tile of destination
    if (br < bc) return;

    __shared__ float tile[32][33];
    const int R0 = br * 32;
    const int C0 = bc * 32;
    const int tx = threadIdx.x;

    // load source tile: rows C0.., cols R0.. (upper region), coalesced in tx
    for (int yy = threadIdx.y; yy < 32; yy += 8) {
        const int sr = C0 + yy;
        const int sc = R0 + tx;
        tile[yy][tx] = (sr < MOUT && sc < MOUT) ? out[(size_t)sr * MOUT + sc] : 0.0f;
    }
    __syncthreads();

    // write transposed, strictly-lower only, coalesced in tx
    for (int yy = threadIdx.y; yy < 32; yy += 8) {
        const int dr = R0 + yy;
        const int dc = C0 + tx;
        if (dr < MOUT && dc < MOUT && dr > dc) {
            out[(size_t)dr * MOUT + dc] = tile[tx][yy];
        }
    }
}

// ---------------------------------------------------------------------------
extern "C" void kernel_launch(void* const* d_in, const int* in_sizes, int n_in,
                              void* d_out, int out_size, void* d_ws, size_t ws_size,
                              hipStream_t stream) {
    const float* data = (const float*)d_in[0];     // (8192, 128) fp32

    float* gram  = (float*)d_ws;                   // 8192 x 8192 fp32
    float* uu_ws = gram + (size_t)LDG * LDG;       // 8191 fp32

    float* out          = (float*)d_out;
    float* out_uu       = out + (size_t)MOUT * MOUT;
    float* out_kfinal   = out_uu + N_ROWS;
    float* out_kdiagfin = out_kfinal + N_ROWS;

    // 1) Gram matrix (upper-triangle tiles), one wave per 16x16 tile
    gram_wmma_kernel<<<dim3(64, 512), 256, 0, stream>>>(data, gram);

    // 2) Sequential diagonal scan -> uu, k_diag_final
    diag_scan_kernel<<<1, 32, 0, stream>>>(gram, uu_ws, out_uu, out_kdiagfin);

    // 3) Per-diagonal recurrence (8191 independent chains)
    diag_chain_kernel<<<32, 256, 0, stream>>>(gram, uu_ws, out, out_kfinal);

    // 4) Mirror lower triangle of k_train
    const int nt = (MOUT + 31) / 32; // 253
    mirror_kernel<<<dim3(nt, nt), dim3(32, 8), 0, stream>>>(out);
}